// Add_Square_2654289789550
// MI455X (gfx1250) — compile-verified
//
#include <hip/hip_runtime.h>
#include <stdint.h>

// ---- problem constants (match reference) ----
#define HH     224
#define NPIX   (HH*HH)        // 50176
#define CQ     3
#define NPC    (CQ*NPIX)      // 150528
#define NQ     5000
#define NINIT  (CQ*HH)        // 672
#define EPSV   0.05f
#define NCHUNK 10
#define CHUNK  (NQ/NCHUNK)    // 500

typedef unsigned int u32x4 __attribute__((ext_vector_type(4)));
typedef unsigned int u32x8 __attribute__((ext_vector_type(8)));

// ---------------- JAX threefry2x32 (exact) ----------------
__device__ __forceinline__ uint32_t rotl32(uint32_t v, int r) {
  return (v << r) | (v >> (32 - r));
}

__device__ void tf2x32(uint32_t k0, uint32_t k1, uint32_t x0, uint32_t x1,
                       uint32_t& o0, uint32_t& o1) {
  uint32_t ks2 = k0 ^ k1 ^ 0x1BD11BDAu;
  x0 += k0; x1 += k1;
#define TFR(r) { x0 += x1; x1 = rotl32(x1, r); x1 ^= x0; }
  TFR(13) TFR(15) TFR(26) TFR(6)   x0 += k1;  x1 += ks2 + 1u;
  TFR(17) TFR(29) TFR(16) TFR(24)  x0 += ks2; x1 += k0 + 2u;
  TFR(13) TFR(15) TFR(26) TFR(6)   x0 += k0;  x1 += k1 + 3u;
  TFR(17) TFR(29) TFR(16) TFR(24)  x0 += k1;  x1 += ks2 + 4u;
  TFR(13) TFR(15) TFR(26) TFR(6)   x0 += ks2; x1 += k0 + 5u;
#undef TFR
  o0 = x0; o1 = x1;
}

__device__ __forceinline__ float bits2u(uint32_t b) {      // uniform [0,1)
  return __uint_as_float((b >> 9) | 0x3f800000u) - 1.0f;
}
// sign(2u-1): 2 => +1, 1 => -1, 0 => sign==0 (no-op)
__device__ __forceinline__ uint32_t signcode(uint32_t bits) {
  float f = 2.0f * bits2u(bits) - 1.0f;
  return f > 0.0f ? 2u : (f < 0.0f ? 1u : 0u);
}

__device__ int side_of(int it) {   // static schedule, exact double math + rint
  double p;
  if      (it > 8000) p = 0.8/512.0;
  else if (it > 6000) p = 0.8/256.0;
  else if (it > 4000) p = 0.8/128.0;
  else if (it > 2000) p = 0.8/64.0;
  else if (it > 1000) p = 0.8/32.0;
  else if (it > 500)  p = 0.8/16.0;
  else if (it > 200)  p = 0.8/8.0;
  else if (it > 50)   p = 0.8/4.0;
  else if (it > 10)   p = 0.8/2.0;
  else                p = 0.8;
  int s = (int)rint(sqrt(p * (double)(CQ*HH*HH) / (double)CQ));
  return s < 1 ? 1 : s;
}

// ---------------- kernel 1: zero atomic buffer + build tables ----------------
__global__ __launch_bounds__(256)
void prep_kernel(uint32_t* __restrict__ last_buf,
                 uint32_t* __restrict__ iter_tab,
                 uint32_t* __restrict__ init_tab) {
  int id = blockIdx.x * blockDim.x + threadIdx.x;
  if (id < NPC) { last_buf[id] = 0u; return; }
  id -= NPC;
  if (id < NQ) {
    const int t = id;
    uint32_t a, b;                         // k = fold_in(base=(0,1), t+1)
    tf2x32(0u, 1u, 0u, (uint32_t)(t + 1), a, b);
    uint32_t w00, w01, w10, w11;           // split(k): counts [0,1,2,3]
    tf2x32(a, b, 0u, 2u, w00, w01);
    tf2x32(a, b, 1u, 3u, w10, w11);
    // k1 = (w00, w10) -> scalar uniform -> vh
    uint32_t r0, r1;
    tf2x32(w00, w10, 0u, 0u, r0, r1);
    const int s  = side_of(t);
    const int vh = (int)floorf(bits2u(r0) * (float)(HH - s));
    // k2 = (w01, w11) -> 3 uniforms (odd-size pad: blocks (0,2),(1,0))
    uint32_t b00, b01, b10, b11;
    tf2x32(w01, w11, 0u, 2u, b00, b01);
    tf2x32(w01, w11, 1u, 0u, b10, b11);
    const uint32_t c0 = signcode(b00);     // bits order: [blk0.w0, blk1.w0, blk0.w1]
    const uint32_t c1 = signcode(b10);
    const uint32_t c2 = signcode(b01);
    iter_tab[t] = (uint32_t)vh | ((uint32_t)s << 8) |
                  (c0 << 16) | (c1 << 18) | (c2 << 20);
    return;
  }
  id -= NQ;
  if (id < NINIT) {                        // init stripe signs: uniform(k0,(1,3,1,224))
    uint32_t a, b;
    tf2x32(0u, 1u, 0u, 0u, a, b);          // k0 = fold_in(base, 0)
    uint32_t w0, w1, bits;
    if (id < NINIT/2) { tf2x32(a, b, (uint32_t)id, (uint32_t)(id + NINIT/2), w0, w1); bits = w0; }
    else              { tf2x32(a, b, (uint32_t)(id - NINIT/2), (uint32_t)id, w0, w1); bits = w1; }
    init_tab[id] = signcode(bits);
  }
}

// ---------------- kernel 2: chunked coverage scan (TDM-staged table) ----------
// grid = (NPIX/256, NCHUNK). Wave 0 of each block issues ONE tensor_load_to_lds
// (Tensor Data Mover) moving its 500-entry iteration-table window into LDS,
// waits on TENSORcnt, then all threads scan one pixel's coverage and commit the
// last non-zero sign via atomicMax on a ((chunk+1)<<2|code) packing.
__global__ __launch_bounds__(256)
void scan_kernel(const uint32_t* __restrict__ iter_tab,
                 uint32_t* __restrict__ last_buf) {
  __shared__ alignas(16) uint32_t tab[CHUNK];
  const int chunk = blockIdx.y;
  const int tid   = threadIdx.x;

  if (tid < 32) {                                    // one wave issues the TDM op
    const uint64_t ga  = (uint64_t)(uintptr_t)(iter_tab + chunk * CHUNK);
    const uint32_t lds = (uint32_t)(uintptr_t)(&tab[0]);
    // D# group 0: count=1 | lds_addr | global_addr[31:0] | ga[56:32], type=2
    const u32x4 g0 = { 1u,
                       lds,
                       (uint32_t)ga,
                       (uint32_t)((ga >> 32) & 0x01ffffffu) | 0x80000000u };
    // D# group 1: data_size=4B; tensor_dim0=CHUNK; tensor_dim1=1;
    //             tile_dim0=CHUNK; tile_dim1=1; dim0_stride=CHUNK
    const u32x8 g1 = { 0x00020000u,
                       ((uint32_t)CHUNK & 0xffffu) << 16,
                       1u << 16,
                       ((uint32_t)CHUNK & 0xffffu) << 16,
                       1u,
                       (uint32_t)CHUNK,
                       0u,
                       0u };
    asm volatile("tensor_load_to_lds %0, %1" :: "s"(g0), "s"(g1) : "memory");
    __builtin_amdgcn_s_wait_tensorcnt(0);
  }
  __syncthreads();

  const int pix = blockIdx.x * blockDim.x + tid;    // 0..NPIX-1
  const int i = pix / HH, j = pix % HH;
  const uint32_t lo = (uint32_t)min(i, j);
  const uint32_t hi = (uint32_t)max(i, j);

  uint32_t c0 = 0u, c1 = 0u, c2 = 0u;
#pragma unroll 4
  for (int t = 0; t < CHUNK; ++t) {
    const uint32_t w  = tab[t];
    const uint32_t vh = w & 0xffu;
    const uint32_t s  = (w >> 8) & 0xffu;
    if (lo >= vh && hi < vh + s) {                  // diagonal square covers pixel
      const uint32_t t0 = (w >> 16) & 3u;
      const uint32_t t1 = (w >> 18) & 3u;
      const uint32_t t2 = (w >> 20) & 3u;
      if (t0) c0 = t0;                              // sign==0 is a true no-op
      if (t1) c1 = t1;
      if (t2) c2 = t2;
    }
  }
  const uint32_t tag = (uint32_t)(chunk + 1) << 2;  // later chunk always wins
  if (c0) atomicMax(&last_buf[0*NPIX + pix], tag | c0);
  if (c1) atomicMax(&last_buf[1*NPIX + pix], tag | c1);
  if (c2) atomicMax(&last_buf[2*NPIX + pix], tag | c2);
}

// ---------------- kernel 3: materialize output ------------------------------
// Init-sign table (broadcast-read) staged into LDS via async b128 loads.
__global__ __launch_bounds__(256)
void final_kernel(const float* __restrict__ x,
                  const uint32_t* __restrict__ last_buf,
                  const uint32_t* __restrict__ init_tab,
                  float* __restrict__ out) {
  __shared__ alignas(16) uint32_t itab[NINIT];
  const int tid = threadIdx.x;
  if (tid < NINIT / 4) {                            // 168 lanes x 16B
    const uint64_t ga = (uint64_t)(uintptr_t)(init_tab + tid * 4);
    const uint32_t la = (uint32_t)(uintptr_t)(&itab[tid * 4]);
    asm volatile("global_load_async_to_lds_b128 %0, %1, off"
                 :: "v"(la), "v"(ga) : "memory");
  }
  asm volatile("s_wait_asynccnt 0" ::: "memory");
  __syncthreads();

  const int id = blockIdx.x * blockDim.x + tid;
  if (id >= NPC) return;
  const int c = id / NPIX;
  const int j = (id % NPIX) % HH;
  const uint32_t packed = last_buf[id];
  const uint32_t code   = packed ? (packed & 3u) : itab[c*HH + j];
  const float xv = x[id];
  const float v  = (code == 2u) ? fminf(xv + EPSV, 1.0f)
                 : (code == 1u) ? fmaxf(xv - EPSV, 0.0f)
                 : xv;
  out[id] = v;
}

// ---------------- entry ----------------
extern "C" void kernel_launch(void* const* d_in, const int* in_sizes, int n_in,
                              void* d_out, int out_size, void* d_ws, size_t ws_size,
                              hipStream_t stream) {
  (void)in_sizes; (void)n_in; (void)out_size; (void)ws_size;
  const float* x   = (const float*)d_in[0];
  float*       out = (float*)d_out;
  uint32_t*    ws  = (uint32_t*)d_ws;

  uint32_t* last_buf = ws;                 // NPC u32 (atomicMax buffer)
  uint32_t* iter_tab = ws + NPC;           // NQ u32
  uint32_t* init_tab = ws + NPC + NQ;      // NINIT u32 (16B-aligned: NPC+NQ % 4 == 0)

  const int total = NPC + NQ + NINIT;
  prep_kernel<<<(total + 255) / 256, 256, 0, stream>>>(last_buf, iter_tab, init_tab);

  dim3 grid(NPIX / 256, NCHUNK);
  scan_kernel<<<grid, 256, 0, stream>>>(iter_tab, last_buf);

  final_kernel<<<(NPC + 255) / 256, 256, 0, stream>>>(x, last_buf, init_tab, out);
}